// LearnedRouter_82257213653414
// MI455X (gfx1250) — compile-verified
//
#include <hip/hip_runtime.h>
#include <hip/hip_bf16.h>

#define Zdim 512
#define Hdim 256
#define XS_STRIDE 516   // 516 % 64 == 4 -> conflict-free A-frag column reads
#define TS_STRIDE 260   // 260 % 64 == 4 -> conflict-free
#define QA_STRIDE 257   // 257 % 64 == 1 -> conflict-free per-lane row reads

typedef __attribute__((ext_vector_type(2))) float v2f;
typedef __attribute__((ext_vector_type(8))) float v8f;

__device__ __forceinline__ float gelu_exact(float x) {
    return 0.5f * x * (1.0f + erff(x * 0.70710678118654752f));
}

// One block = 4 waves = 128 threads, handles 16 rows of X.
// out = (LayerNorm(gelu(X @ W + bias)) ) @ W1[w1_row_off : w1_row_off+256, :]
__global__ __launch_bounds__(128)
void transform_kernel(const float* __restrict__ X, const float* __restrict__ W,
                      const float* __restrict__ bias,
                      const float* __restrict__ lnw, const float* __restrict__ lnb,
                      const float* __restrict__ W1, int w1_row_off,
                      float* __restrict__ outA)
{
    __shared__ float xs[16 * XS_STRIDE];
    __shared__ float ts[16 * TS_STRIDE];
    __shared__ float red[16 * 8 * 2];
    __shared__ float mr[16 * 2];

    const int tid  = threadIdx.x;
    const int row0 = blockIdx.x * 16;

    // Stage 16 x 512 input rows into LDS.
    for (int i = tid; i < 16 * Zdim; i += 128) {
        int r = i >> 9, c = i & 511;
        xs[r * XS_STRIDE + c] = X[(size_t)(row0 + r) * Zdim + c];
    }
    __syncthreads();

    const int wid  = tid >> 5;
    const int lane = tid & 31;
    const int half = lane >> 4;      // 0: lanes 0-15, 1: lanes 16-31
    const int l16  = lane & 15;      // M index for A, N index for B/C/D
    const int koff = half * 2;       // K sub-offset per ISA 16x4 f32 A layout

    // ---- GEMM1 (K=512) + bias + GELU -> ts ----
    for (int t = 0; t < 4; ++t) {
        const int col0 = (wid + 4 * t) * 16;
        v8f acc = {};
        #pragma unroll 4
        for (int k = 0; k < Zdim; k += 4) {
            v2f a, b;
            a.x = xs[l16 * XS_STRIDE + k + koff];
            a.y = xs[l16 * XS_STRIDE + k + koff + 1];
            b.x = W[(size_t)(k + koff) * Hdim + col0 + l16];
            b.y = W[(size_t)(k + koff + 1) * Hdim + col0 + l16];
            acc = __builtin_amdgcn_wmma_f32_16x16x4_f32(false, a, false, b,
                                                        (short)0, acc, false, false);
        }
        const float bv = bias[col0 + l16];
        #pragma unroll
        for (int r = 0; r < 8; ++r) {
            int m = r + half * 8;
            ts[m * TS_STRIDE + col0 + l16] = gelu_exact(acc[r] + bv);
        }
    }
    __syncthreads();

    // ---- LayerNorm over H=256 per row (16 rows, 8 threads each) ----
    {
        int r = tid >> 3, sub = tid & 7;
        float s = 0.f, ss = 0.f;
        for (int c = sub * 32; c < sub * 32 + 32; ++c) {
            float v = ts[r * TS_STRIDE + c];
            s += v; ss += v * v;
        }
        red[(r * 8 + sub) * 2 + 0] = s;
        red[(r * 8 + sub) * 2 + 1] = ss;
    }
    __syncthreads();
    if (tid < 16) {
        float s = 0.f, ss = 0.f;
        for (int j = 0; j < 8; ++j) {
            s  += red[(tid * 8 + j) * 2 + 0];
            ss += red[(tid * 8 + j) * 2 + 1];
        }
        float mean = s * (1.0f / Hdim);
        float var  = ss * (1.0f / Hdim) - mean * mean;
        mr[tid * 2 + 0] = mean;
        mr[tid * 2 + 1] = rsqrtf(var + 1e-5f);
    }
    __syncthreads();
    for (int i = tid; i < 16 * Hdim; i += 128) {
        int r = i >> 8, c = i & 255;
        float v = ts[r * TS_STRIDE + c];
        ts[r * TS_STRIDE + c] = (v - mr[r * 2]) * mr[r * 2 + 1] * lnw[c] + lnb[c];
    }
    __syncthreads();

    // ---- GEMM2 (K=256): ts @ W1[w1_row_off + k][:] -> outA ----
    for (int t = 0; t < 4; ++t) {
        const int col0 = (wid + 4 * t) * 16;
        v8f acc = {};
        #pragma unroll 4
        for (int k = 0; k < Hdim; k += 4) {
            v2f a, b;
            a.x = ts[l16 * TS_STRIDE + k + koff];
            a.y = ts[l16 * TS_STRIDE + k + koff + 1];
            b.x = W1[(size_t)(w1_row_off + k + koff) * Hdim + col0 + l16];
            b.y = W1[(size_t)(w1_row_off + k + koff + 1) * Hdim + col0 + l16];
            acc = __builtin_amdgcn_wmma_f32_16x16x4_f32(false, a, false, b,
                                                        (short)0, acc, false, false);
        }
        #pragma unroll
        for (int r = 0; r < 8; ++r) {
            int m = r + half * 8;
            outA[(size_t)(row0 + m) * Hdim + col0 + l16] = acc[r];
        }
    }
}

// scores[b,n] = (b2 + sum_h gelu(qa[b,h]+da[n,h]+b1[h]) * W2[h]) / (|T|+eps)
// lane == batch index b; one doc per wave-iteration.
__global__ __launch_bounds__(256)
void score_kernel(const float* __restrict__ qa, const float* __restrict__ da,
                  const float* __restrict__ b1, const float* __restrict__ W2,
                  const float* __restrict__ b2, const float* __restrict__ temp,
                  int nB, int nN, float* __restrict__ scores)
{
    __shared__ float qa_s[32 * QA_STRIDE];
    __shared__ float b1s[Hdim];
    __shared__ float w2s[Hdim];

    const int tid = threadIdx.x;
    for (int i = tid; i < nB * Hdim; i += 256)
        qa_s[(i >> 8) * QA_STRIDE + (i & 255)] = qa[i];
    if (tid < Hdim) { b1s[tid] = b1[tid]; w2s[tid] = W2[tid]; }
    __syncthreads();

    const float inv_t = 1.0f / (fabsf(temp[0]) + 1e-8f);
    const float b2v   = b2[0];
    const int lane    = tid & 31;
    const int gwave   = blockIdx.x * 8 + (tid >> 5);
    const int nwaves  = gridDim.x * 8;
    const float* qp   = qa_s + lane * QA_STRIDE;

    for (int n = gwave; n < nN; n += nwaves) {
        const float* dp = da + (size_t)n * Hdim;
        float s = 0.f;
        #pragma unroll 4
        for (int h = 0; h < Hdim; h += 4) {
            float4 dv = *(const float4*)(dp + h);
            s += gelu_exact(qp[h + 0] + dv.x + b1s[h + 0]) * w2s[h + 0];
            s += gelu_exact(qp[h + 1] + dv.y + b1s[h + 1]) * w2s[h + 1];
            s += gelu_exact(qp[h + 2] + dv.z + b1s[h + 2]) * w2s[h + 2];
            s += gelu_exact(qp[h + 3] + dv.w + b1s[h + 3]) * w2s[h + 3];
        }
        if (lane < nB)
            scores[(size_t)lane * nN + n] = (s + b2v) * inv_t;
    }
}

// One block per batch row: top-k over nN, then softmax over the k winners.
// Output layout: out[0 .. nB*k) = indices (as float), out[nB*k .. 2*nB*k) = probs.
__global__ __launch_bounds__(256)
void topk_kernel(const float* __restrict__ scores, const int* __restrict__ kptr,
                 int nB, int nN, float* __restrict__ out)
{
    __shared__ float cv[256 * 5];
    __shared__ int   ci[256 * 5];

    const int b   = blockIdx.x;
    const int tid = threadIdx.x;
    int k = kptr[0];
    if (k > 5) k = 5;
    if (k < 1) k = 1;

    float bv[5]; int bi[5];
    #pragma unroll
    for (int i = 0; i < 5; ++i) { bv[i] = -3.4e38f; bi[i] = -1; }

    const float* row = scores + (size_t)b * nN;
    for (int n = tid; n < nN; n += 256) {
        float v = row[n];
        if (v > bv[4]) {
            int pos = 4;
            while (pos > 0 && v > bv[pos - 1]) {
                bv[pos] = bv[pos - 1]; bi[pos] = bi[pos - 1]; --pos;
            }
            bv[pos] = v; bi[pos] = n;
        }
    }
    #pragma unroll
    for (int i = 0; i < 5; ++i) { cv[tid * 5 + i] = bv[i]; ci[tid * 5 + i] = bi[i]; }
    __syncthreads();

    if (tid == 0) {
        float sel_v[5]; int sel_i[5];
        for (int r = 0; r < k; ++r) {
            float mv = -3.4e38f; int mp = 0;
            for (int j = 0; j < 256 * 5; ++j)
                if (cv[j] > mv) { mv = cv[j]; mp = j; }
            sel_v[r] = mv; sel_i[r] = ci[mp];
            cv[mp] = -3.4e38f;
        }
        float mx = sel_v[0];     // sorted descending by construction
        float e[5]; float sum = 0.f;
        for (int r = 0; r < k; ++r) { e[r] = expf(sel_v[r] - mx); sum += e[r]; }
        for (int r = 0; r < k; ++r) {
            out[b * k + r] = (float)sel_i[r];
            out[(size_t)nB * k + b * k + r] = e[r] / sum;
        }
    }
}

extern "C" void kernel_launch(void* const* d_in, const int* in_sizes, int n_in,
                              void* d_out, int out_size, void* d_ws, size_t ws_size,
                              hipStream_t stream)
{
    const float* q    = (const float*)d_in[0];
    const float* docs = (const float*)d_in[1];
    const int*   kp   = (const int*)d_in[2];
    const float* Wq   = (const float*)d_in[3];
    const float* bq   = (const float*)d_in[4];
    const float* lnqw = (const float*)d_in[5];
    const float* lnqb = (const float*)d_in[6];
    const float* Wd   = (const float*)d_in[7];
    const float* bd   = (const float*)d_in[8];
    const float* lndw = (const float*)d_in[9];
    const float* lndb = (const float*)d_in[10];
    const float* W1   = (const float*)d_in[11];
    const float* b1   = (const float*)d_in[12];
    const float* W2   = (const float*)d_in[13];
    const float* b2   = (const float*)d_in[14];
    const float* temp = (const float*)d_in[15];

    const int nB = in_sizes[0] / Zdim;   // 32
    const int nN = in_sizes[1] / Zdim;   // 20000

    float* qa = (float*)d_ws;                       // [nB, 256]
    float* da = qa + (size_t)nB * Hdim;             // [nN, 256]
    float* sc = da + (size_t)nN * Hdim;             // [nB, nN]

    transform_kernel<<<nB / 16, 128, 0, stream>>>(q,    Wq, bq, lnqw, lnqb, W1, 0,    qa);
    transform_kernel<<<nN / 16, 128, 0, stream>>>(docs, Wd, bd, lndw, lndb, W1, Hdim, da);
    score_kernel<<<512, 256, 0, stream>>>(qa, da, b1, W2, b2, temp, nB, nN, sc);
    topk_kernel<<<nB, 256, 0, stream>>>(sc, kp, nB, nN, (float*)d_out);
}